// GNNembedding_58815282151629
// MI455X (gfx1250) — compile-verified
//
#include <hip/hip_runtime.h>
#include <hip/hip_bf16.h>

// -------------------------------------------------------------------------
// 2-layer GraphSAGE for MI455X (gfx1250, wave32, WMMA).
//   h   = (agg1/deg) @ Wl1^T + bl1 + x @ Wr1^T       (D_IN=128 -> D_HID=256)
//   out = (agg2/deg) @ Wl2^T + bl2 + h @ Wr2^T       (D_HID=256 -> D_OUT=128)
// agg = scatter-sum of source rows over edges, deg = in-degree.
// GEMMs run on V_WMMA_F32_16X16X4_F32 (exact f32, compute is tiny vs the
// L2-resident gather/scatter traffic which dominates on a 23.3 TB/s part).
// -------------------------------------------------------------------------

typedef __attribute__((ext_vector_type(2))) float v2f;
typedef __attribute__((ext_vector_type(8))) float v8f;

// ---------------- workspace zeroing (deg + agg1 + agg2) -------------------
__global__ void sage_zero_kernel(float4* __restrict__ p, long n4) {
    long i = (long)blockIdx.x * blockDim.x + threadIdx.x;
    long stride = (long)gridDim.x * blockDim.x;
    float4 z; z.x = 0.f; z.y = 0.f; z.z = 0.f; z.w = 0.f;
    for (; i < n4; i += stride) p[i] = z;
}

// ---------------- degree: deg[dst] += 1 -----------------------------------
__global__ void sage_deg_kernel(const long long* __restrict__ ei,
                                float* __restrict__ deg, int E) {
    int e = blockIdx.x * blockDim.x + threadIdx.x;
    if (e < E) {
        int d = (int)ei[E + e];           // row 1 = dst
        atomicAdd(&deg[d], 1.0f);
    }
}

// ---------------- edge scatter: agg[dst,:] += x[src,:] --------------------
// One thread per (edge, 4-float chunk): coalesced b128 row loads from L2,
// 4x global_atomic_add_f32 into the accumulation buffer.
template <int D>
__global__ void sage_scatter_kernel(const float* __restrict__ x,
                                    const long long* __restrict__ ei,
                                    float* __restrict__ agg, int E) {
    const int CH = D / 4;
    long gid = (long)blockIdx.x * blockDim.x + threadIdx.x;
    long total = (long)E * CH;
    if (gid >= total) return;
    int e = (int)(gid / CH);
    int c = (int)(gid % CH) * 4;
    int s = (int)ei[e];                   // row 0 = src
    int d = (int)ei[E + e];               // row 1 = dst
    const float4 v = *(const float4*)&x[(long)s * D + c];
    float* dst = &agg[(long)d * D + c];
    atomicAdd(dst + 0, v.x);
    atomicAdd(dst + 1, v.y);
    atomicAdd(dst + 2, v.z);
    atomicAdd(dst + 3, v.w);
}

// ---------------- fused SAGE linear layer on WMMA f32 ---------------------
// Block = 256 threads = 8 wave32s, owns a 16-node strip.
// LDS holds the degree-normalized agg tile and the self-feature tile
// (16 x K f32 each). Each wave produces DOUT/16/8 16x16 output tiles,
// accumulating BOTH matmuls into one accumulator seeded with the bias.
//
// V_WMMA_F32_16X16X4_F32 operand layout (wave32):
//   A (16x4): lane = M + 16*g, vgpr v  ->  A[M][2*g + v]   (float2 per lane)
//   B (4x16): lane = N + 16*g, vgpr v  ->  B[2*g + v][N]   == W[N][2*g+v]
//   C/D     : lane = N + 16*g, vgpr r  ->  D[r + 8*g][N]
template <int K, int DOUT>
__global__ __launch_bounds__(256) void sage_wmma_kernel(
    const float* __restrict__ in,    // [N,K]   self features
    const float* __restrict__ agg,   // [N,K]   neighbor sums
    const float* __restrict__ deg,   // [N]
    const float* __restrict__ Wl,    // [DOUT,K]
    const float* __restrict__ bl,    // [DOUT]
    const float* __restrict__ Wr,    // [DOUT,K]
    float* __restrict__ out)         // [N,DOUT]
{
    __shared__ float sAgg[16][K];
    __shared__ float sIn[16][K];

    const int node0 = blockIdx.x * 16;
    const int tid = threadIdx.x;

    // Stage the 16-node strip into LDS, folding in the 1/max(deg,1) scale.
    for (int i = tid; i < 16 * K / 4; i += 256) {
        int r = (i * 4) / K;
        int c = (i * 4) % K;
        const float4 vx = *(const float4*)&in[(long)(node0 + r) * K + c];
        *(float4*)&sIn[r][c] = vx;
        float inv = 1.0f / fmaxf(deg[node0 + r], 1.0f);
        float4 va = *(const float4*)&agg[(long)(node0 + r) * K + c];
        va.x *= inv; va.y *= inv; va.z *= inv; va.w *= inv;
        *(float4*)&sAgg[r][c] = va;
    }
    __syncthreads();

    const int wave = tid >> 5;        // wave32
    const int lane = tid & 31;
    const int mrow = lane & 15;       // M (for A) / N (for B, C/D)
    const int gk   = lane >> 4;       // K-group (A/B) / M-half (C/D)

    constexpr int FT  = DOUT / 16;    // feature tiles (16 or 8)
    constexpr int TPW = FT / 8;       // tiles per wave (2 or 1)

    for (int t = 0; t < TPW; ++t) {
        const int feat0 = (wave * TPW + t) * 16;

        // Seed accumulator with bias: D[m][n] starts at bl[feat0+n].
        float bv = bl[feat0 + mrow];
        v8f acc = {bv, bv, bv, bv, bv, bv, bv, bv};

        for (int k = 0; k < K; k += 4) {
            const int kc = k + 2 * gk;
            v2f aAgg = *(const v2f*)&sAgg[mrow][kc];
            v2f aIn  = *(const v2f*)&sIn[mrow][kc];
            v2f bL   = *(const v2f*)&Wl[(long)(feat0 + mrow) * K + kc];
            v2f bR   = *(const v2f*)&Wr[(long)(feat0 + mrow) * K + kc];
            acc = __builtin_amdgcn_wmma_f32_16x16x4_f32(
                false, aAgg, false, bL, (short)0, acc, false, false);
            acc = __builtin_amdgcn_wmma_f32_16x16x4_f32(
                false, aIn,  false, bR, (short)0, acc, false, false);
        }

#pragma unroll
        for (int r = 0; r < 8; ++r) {
            int m = r + 8 * gk;
            out[(long)(node0 + m) * DOUT + feat0 + mrow] = acc[r];
        }
    }
}

// --------------------------------------------------------------------------
extern "C" void kernel_launch(void* const* d_in, const int* in_sizes, int n_in,
                              void* d_out, int out_size, void* d_ws, size_t ws_size,
                              hipStream_t stream) {
    const float*     x   = (const float*)d_in[0];
    const long long* ei  = (const long long*)d_in[1];   // int64 per reference
    const float*     Wl1 = (const float*)d_in[2];
    const float*     bl1 = (const float*)d_in[3];
    const float*     Wr1 = (const float*)d_in[4];
    const float*     Wl2 = (const float*)d_in[5];
    const float*     bl2 = (const float*)d_in[6];
    const float*     Wr2 = (const float*)d_in[7];
    float* out = (float*)d_out;

    constexpr int DIN = 128, DHID = 256;
    const int N = in_sizes[0] / DIN;     // 50000 (divisible by 16)
    const int E = in_sizes[1] / 2;       // 800000

    // Workspace layout (floats): [deg N][agg1 N*128][agg2 N*256][h N*256]
    float* ws   = (float*)d_ws;
    float* deg  = ws;
    float* agg1 = ws + N;
    float* agg2 = agg1 + (size_t)N * DIN;
    float* h    = agg2 + (size_t)N * DHID;

    // 1) zero deg + agg1 + agg2 (h and out are fully overwritten)
    long zero4 = (long)N * (1 + DIN + DHID) / 4;
    sage_zero_kernel<<<2048, 256, 0, stream>>>((float4*)ws, zero4);

    // 2) in-degrees
    sage_deg_kernel<<<(E + 255) / 256, 256, 0, stream>>>(ei, deg, E);

    // 3) layer-1 scatter: agg1[dst] += x[src]
    {
        long total = (long)E * (DIN / 4);
        sage_scatter_kernel<DIN><<<(int)((total + 255) / 256), 256, 0, stream>>>(
            x, ei, agg1, E);
    }

    // 4) h = (agg1/deg) @ Wl1^T + bl1 + x @ Wr1^T
    sage_wmma_kernel<DIN, DHID><<<N / 16, 256, 0, stream>>>(
        x, agg1, deg, Wl1, bl1, Wr1, h);

    // 5) layer-2 scatter: agg2[dst] += h[src]
    {
        long total = (long)E * (DHID / 4);
        sage_scatter_kernel<DHID><<<(int)((total + 255) / 256), 256, 0, stream>>>(
            h, ei, agg2, E);
    }

    // 6) out = (agg2/deg) @ Wl2^T + bl2 + h @ Wr2^T
    sage_wmma_kernel<DHID, DIN><<<N / 16, 256, 0, stream>>>(
        h, agg2, deg, Wl2, bl2, Wr2, out);
}